// NollaFraud_28630251995136
// MI455X (gfx1250) — compile-verified
//
#include <hip/hip_runtime.h>
#include <hip/hip_bf16.h>
#include <math.h>

typedef __attribute__((ext_vector_type(2))) float v2f;
typedef __attribute__((ext_vector_type(8))) float v8f;

// ---------------------------------------------------------------------------
// Softmax of alpha1 [2d,3] and alpha2 [4d,3] -> W1s, W2s
// ---------------------------------------------------------------------------
__global__ void softmax_weights_kernel(const float* __restrict__ a1,
                                       const float* __restrict__ a2,
                                       float* __restrict__ w1s,
                                       float* __restrict__ w2s,
                                       int d) {
    int idx = blockIdx.x * blockDim.x + threadIdx.x;
    int n1 = 2 * d;
    int n2 = 4 * d;
    const float* src;
    float* dst;
    int row;
    if (idx < n1) {
        src = a1; dst = w1s; row = idx;
    } else if (idx < n1 + n2) {
        src = a2; dst = w2s; row = idx - n1;
    } else {
        return;
    }
    float x0 = src[row * 3 + 0];
    float x1 = src[row * 3 + 1];
    float x2 = src[row * 3 + 2];
    float m  = fmaxf(x0, fmaxf(x1, x2));
    float e0 = __expf(x0 - m);
    float e1 = __expf(x1 - m);
    float e2 = __expf(x2 - m);
    float inv = 1.0f / (e0 + e1 + e2);
    dst[row * 3 + 0] = e0 * inv;
    dst[row * 3 + 1] = e1 * inv;
    dst[row * 3 + 2] = e2 * inv;
}

// ---------------------------------------------------------------------------
// h0 = feat[N,F] @ Wm[F,d] via V_WMMA_F32_16X16X4_F32 (fp32, full precision).
// One wave per 16x16 tile; masked loads keep EXEC all-ones at the WMMA.
// ---------------------------------------------------------------------------
__global__ __launch_bounds__(32)
void mlp_wmma_kernel(const float* __restrict__ feat,
                     const float* __restrict__ Wm,
                     float* __restrict__ h0,
                     int N, int F, int d) {
    const int lane = threadIdx.x;
    const int r    = lane & 15;
    const int hi   = lane >> 4;          // 0 or 1
    const int row0 = blockIdx.x * 16;
    const int col0 = blockIdx.y * 16;

    const int rowA  = row0 + r;
    const int rowC  = rowA < N ? rowA : N - 1;
    const float mR  = rowA < N ? 1.0f : 0.0f;

    v8f acc = {0.0f, 0.0f, 0.0f, 0.0f, 0.0f, 0.0f, 0.0f, 0.0f};

    for (int k0 = 0; k0 < F; k0 += 4) {
        const int ka = k0 + hi * 2;
        const int kb = ka + 1;
        const int kac = ka < F ? ka : 0;
        const int kbc = kb < F ? kb : 0;
        const float mA = (ka < F) ? 1.0f : 0.0f;
        const float mB = (kb < F) ? 1.0f : 0.0f;

        v2f a, b;
        a.x = feat[(size_t)rowC * F + kac] * (mA * mR);
        a.y = feat[(size_t)rowC * F + kbc] * (mB * mR);
        b.x = Wm[(size_t)kac * d + col0 + r] * mA;
        b.y = Wm[(size_t)kbc * d + col0 + r] * mB;

        acc = __builtin_amdgcn_wmma_f32_16x16x4_f32(
            false, a, false, b, (short)0, acc, false, false);
    }

#pragma unroll
    for (int v = 0; v < 8; ++v) {
        int row = row0 + v + hi * 8;
        if (row < N) h0[(size_t)row * d + col0 + r] = acc[v];
    }
}

// ---------------------------------------------------------------------------
// Layer-1 InterAgg, wave-per-node fast path (requires d == 64, K <= 32).
// Lane L owns features {2L, 2L+1}. Neighbor indices are loaded once per wave
// (lane k holds nb[k]) and broadcast with __shfl; row gathers are float2
// (global_load_b64, 32 lanes == one full 256B row per instruction).
// ---------------------------------------------------------------------------
__global__ __launch_bounds__(256)
void inter1_wave_kernel(const float* __restrict__ h0,
                        const int* __restrict__ adj,
                        const float* __restrict__ W1s,
                        float* __restrict__ h1,
                        int N, int d, int K) {
    extern __shared__ float sW[];          // 2d*3
    const int tid = threadIdx.x;
    const int nW = 2 * d * 3;
    for (int i = tid; i < nW; i += blockDim.x) sW[i] = W1s[i];
    __syncthreads();

    const int lane = tid & 31;
    const int wave = tid >> 5;
    const int n = blockIdx.x * (blockDim.x >> 5) + wave;
    if (n >= N) return;

    const int h = d >> 1;                  // float2 per row (32 when d=64)
    const float2* h0v = (const float2*)h0;
    const float2 self = h0v[(size_t)n * h + lane];
    const int f0 = 2 * lane;
    const int f1 = f0 + 1;
    const float invK = 1.0f / (float)K;

    float w0x = 0.f, w0y = 0.f, w1x = 0.f, w1y = 0.f;

    for (int rel = 0; rel < 3; ++rel) {
        const int* nb = adj + ((size_t)rel * N + n) * K;
        float sx = 0.f, sy = 0.f;
        for (int k0 = 0; k0 < K; k0 += 32) {
            const int kcnt = (K - k0) < 32 ? (K - k0) : 32;
            const int myIdx = (lane < kcnt) ? nb[k0 + lane] : 0;
#pragma unroll 8
            for (int k = 0; k < kcnt; ++k) {
                const int idx = __shfl(myIdx, k);
                const float2 v = h0v[(size_t)idx * h + lane];
                sx += v.x;
                sy += v.y;
            }
        }
        const float mx = sx * invK;
        const float my = sy * invK;
        w0x += mx * sW[f0 * 3 + rel];
        w0y += my * sW[f1 * 3 + rel];
        w1x += (self.x - mx) * sW[(d + f0) * 3 + rel];
        w1y += (self.y - my) * sW[(d + f1) * 3 + rel];
    }

    float2* h1v = (float2*)h1;
    const size_t o = (size_t)n * 3 * h;
    float2 t;
    h1v[o + lane] = self;
    t.x = w0x; t.y = w0y;
    h1v[o + h + lane] = t;
    t.x = w1x; t.y = w1y;
    h1v[o + 2 * h + lane] = t;
}

// Generic fallback (any d dividing 256, any K)
__global__ __launch_bounds__(256)
void inter1_kernel(const float* __restrict__ h0,
                   const int* __restrict__ adj,
                   const float* __restrict__ W1s,
                   float* __restrict__ h1,
                   int N, int d, int K) {
    extern __shared__ float sW[];
    const int tid = threadIdx.x;
    const int nW = 2 * d * 3;
    for (int i = tid; i < nW; i += blockDim.x) sW[i] = W1s[i];
    __syncthreads();

    const int npb = blockDim.x / d;
    const int n = blockIdx.x * npb + tid / d;
    const int f = tid % d;
    if (n >= N) return;

    const float self = h0[(size_t)n * d + f];
    const float invK = 1.0f / (float)K;
    float w0 = 0.0f, w1 = 0.0f;

    for (int rel = 0; rel < 3; ++rel) {
        const int* nb = adj + ((size_t)rel * N + n) * K;
        float s = 0.0f;
#pragma unroll 8
        for (int k = 0; k < K; ++k) s += h0[(size_t)nb[k] * d + f];
        const float mean = s * invK;
        w0 += mean * sW[f * 3 + rel];
        w1 += (self - mean) * sW[(d + f) * 3 + rel];
    }

    const size_t o = (size_t)n * 3 * d;
    h1[o + f]         = self;
    h1[o + d + f]     = w0;
    h1[o + 2 * d + f] = w1;
}

// ---------------------------------------------------------------------------
// Layer-2 InterAgg, wave-per-node fast path (requires d == 64, K <= 32).
// Tail dim 2d = 128 -> lane L owns features {4L..4L+3}, float4 gathers
// (global_load_b128, 32 lanes == one full 512B tail row per instruction).
// ---------------------------------------------------------------------------
__global__ __launch_bounds__(256)
void inter2_wave_kernel(const float* __restrict__ h1,
                        const int* __restrict__ adj,
                        const int* __restrict__ nodes,
                        const float* __restrict__ W2s,
                        float* __restrict__ h2,
                        int N, int d, int K, int B) {
    extern __shared__ float sW[];          // 4d*3
    const int tid = threadIdx.x;
    const int td = 2 * d;
    const int nW = 2 * td * 3;
    for (int i = tid; i < nW; i += blockDim.x) sW[i] = W2s[i];
    __syncthreads();

    const int lane = tid & 31;
    const int wave = tid >> 5;
    const int i = blockIdx.x * (blockDim.x >> 5) + wave;
    if (i >= B) return;
    const int node = nodes[i];

    const int rowq = (3 * d) >> 2;         // float4 per full h1 row (48)
    const int tailq = d >> 2;              // float4 offset of tail (16)
    const float4* h1v = (const float4*)h1;
    const size_t base4 = (size_t)node * rowq;

    const float4 self = h1v[base4 + tailq + lane];
    const int f0 = 4 * lane;
    const float invK = 1.0f / (float)K;

    float w0x=0.f, w0y=0.f, w0z=0.f, w0w=0.f;
    float w1x=0.f, w1y=0.f, w1z=0.f, w1w=0.f;

    for (int rel = 0; rel < 3; ++rel) {
        const int* nb = adj + ((size_t)rel * N + node) * K;
        float sx=0.f, sy=0.f, sz=0.f, sw=0.f;
        for (int k0 = 0; k0 < K; k0 += 32) {
            const int kcnt = (K - k0) < 32 ? (K - k0) : 32;
            const int myIdx = (lane < kcnt) ? nb[k0 + lane] : 0;
#pragma unroll 8
            for (int k = 0; k < kcnt; ++k) {
                const int idx = __shfl(myIdx, k);
                const float4 v = h1v[(size_t)idx * rowq + tailq + lane];
                sx += v.x; sy += v.y; sz += v.z; sw += v.w;
            }
        }
        const float mx = sx*invK, my = sy*invK, mz = sz*invK, mw = sw*invK;
        w0x += mx * sW[(f0+0)*3 + rel];
        w0y += my * sW[(f0+1)*3 + rel];
        w0z += mz * sW[(f0+2)*3 + rel];
        w0w += mw * sW[(f0+3)*3 + rel];
        w1x += (self.x - mx) * sW[(td+f0+0)*3 + rel];
        w1y += (self.y - my) * sW[(td+f0+1)*3 + rel];
        w1z += (self.z - mz) * sW[(td+f0+2)*3 + rel];
        w1w += (self.w - mw) * sW[(td+f0+3)*3 + rel];
    }

    float4* h2v = (float4*)h2;
    const size_t o4 = (size_t)i * ((7 * d) >> 2);   // 112 float4 per node
    // self_full: 3d floats = rowq float4 copied by 32 lanes (48 = 32 + 16)
    h2v[o4 + lane] = h1v[base4 + lane];
    if (lane < rowq - 32) h2v[o4 + 32 + lane] = h1v[base4 + 32 + lane];
    float4 t;
    t.x = w0x; t.y = w0y; t.z = w0z; t.w = w0w;
    h2v[o4 + rowq + lane] = t;
    t.x = w1x; t.y = w1y; t.z = w1z; t.w = w1w;
    h2v[o4 + rowq + 32 + lane] = t;
}

// Generic fallback
__global__ void inter2_kernel(const float* __restrict__ h1,
                              const int* __restrict__ adj,
                              const int* __restrict__ nodes,
                              const float* __restrict__ W2s,
                              float* __restrict__ h2,
                              int N, int d, int K, int B) {
    extern __shared__ float sW[];
    const int tid = threadIdx.x;
    const int td = 2 * d;
    const int nW = 2 * td * 3;
    for (int i = tid; i < nW; i += blockDim.x) sW[i] = W2s[i];
    __syncthreads();

    const int i = blockIdx.x;
    if (i >= B) return;
    const int node = nodes[i];
    const int f = tid;

    const size_t base = (size_t)node * 3 * d;
    const float self_tail = h1[base + d + f];
    const float invK = 1.0f / (float)K;
    float w0 = 0.0f, w1 = 0.0f;

    for (int rel = 0; rel < 3; ++rel) {
        const int* nb = adj + ((size_t)rel * N + node) * K;
        float s = 0.0f;
#pragma unroll 8
        for (int k = 0; k < K; ++k) s += h1[(size_t)nb[k] * 3 * d + d + f];
        const float mean = s * invK;
        w0 += mean * sW[f * 3 + rel];
        w1 += (self_tail - mean) * sW[(td + f) * 3 + rel];
    }

    const size_t o = (size_t)i * 7 * d;
    h2[o + f] = h1[base + f];
    if (f < d) h2[o + td + f] = h1[base + td + f];
    h2[o + 3 * d + f] = w0;
    h2[o + 5 * d + f] = w1;
}

// ---------------------------------------------------------------------------
// out[i,c] = h2[i,:] . lw[:,c] + log(prior[c]) — one wave per node,
// strided coalesced reads + __shfl_down wave reduction.
// ---------------------------------------------------------------------------
__global__ __launch_bounds__(256)
void final_linear_kernel(const float* __restrict__ h2,
                         const float* __restrict__ lw,
                         const float* __restrict__ prior,
                         float* __restrict__ out,
                         int B, int d) {
    const int lane = threadIdx.x & 31;
    const int wave = threadIdx.x >> 5;
    const int i = blockIdx.x * (blockDim.x >> 5) + wave;
    if (i >= B) return;

    const int H = 7 * d;
    const float* row = h2 + (size_t)i * H;
    float c0 = 0.0f, c1 = 0.0f;
    for (int j = lane; j < H; j += 32) {
        const float v = row[j];
        c0 += v * lw[j * 2 + 0];
        c1 += v * lw[j * 2 + 1];
    }
#pragma unroll
    for (int off = 16; off > 0; off >>= 1) {
        c0 += __shfl_down(c0, off);
        c1 += __shfl_down(c1, off);
    }
    if (lane == 0) {
        out[i * 2 + 0] = c0 + __logf(prior[0]);
        out[i * 2 + 1] = c1 + __logf(prior[1]);
    }
}

// ---------------------------------------------------------------------------
// Launch
// ---------------------------------------------------------------------------
extern "C" void kernel_launch(void* const* d_in, const int* in_sizes, int n_in,
                              void* d_out, int out_size, void* d_ws, size_t ws_size,
                              hipStream_t stream) {
    const float* feat   = (const float*)d_in[0];
    const float* W_mlp  = (const float*)d_in[1];
    const float* alpha1 = (const float*)d_in[2];
    const float* alpha2 = (const float*)d_in[3];
    const float* lin_w  = (const float*)d_in[4];
    const float* prior  = (const float*)d_in[5];
    const int*   adj    = (const int*)d_in[6];
    const int*   nodes  = (const int*)d_in[7];
    float* out = (float*)d_out;

    // Shapes (reference: N=50000, K=32, d=64, B=1024, F=25)
    const int B = in_sizes[7];
    const int d = in_sizes[2] / 6;
    const int F = in_sizes[1] / d;
    const int N = in_sizes[0] / F;
    const int K = in_sizes[6] / (3 * N);

    // Workspace carve-up (floats)
    float* ws  = (float*)d_ws;
    float* w1s = ws;                       // 2d*3
    float* w2s = w1s + (size_t)2 * d * 3;  // 4d*3
    float* h0  = w2s + (size_t)4 * d * 3;  // N*d
    float* h1  = h0  + (size_t)N * d;      // N*3d
    float* h2  = h1  + (size_t)N * 3 * d;  // B*7d

    const bool fast = (d == 64) && (K <= 32);

    // Relation softmax weights
    {
        int total = 6 * d;
        softmax_weights_kernel<<<(total + 255) / 256, 256, 0, stream>>>(
            alpha1, alpha2, w1s, w2s, d);
    }

    // h0 = feat @ W_mlp (fp32 WMMA, one wave per 16x16 tile)
    {
        dim3 grid((N + 15) / 16, d / 16);
        mlp_wmma_kernel<<<grid, 32, 0, stream>>>(feat, W_mlp, h0, N, F, d);
    }

    // Layer-1 InterAgg over all N nodes (dominant gather kernel)
    {
        size_t shm = (size_t)2 * d * 3 * sizeof(float);
        if (fast) {
            const int wpb = 8;             // 8 waves = 8 nodes per block
            int grid = (N + wpb - 1) / wpb;
            inter1_wave_kernel<<<grid, 32 * wpb, shm, stream>>>(
                h0, adj, w1s, h1, N, d, K);
        } else {
            int npb = 256 / d;
            int grid = (N + npb - 1) / npb;
            inter1_kernel<<<grid, 256, shm, stream>>>(h0, adj, w1s, h1, N, d, K);
        }
    }

    // Layer-2 InterAgg over batch nodes
    {
        size_t shm = (size_t)4 * d * 3 * sizeof(float);
        if (fast) {
            const int wpb = 8;
            int grid = (B + wpb - 1) / wpb;
            inter2_wave_kernel<<<grid, 32 * wpb, shm, stream>>>(
                h1, adj, nodes, w2s, h2, N, d, K, B);
        } else {
            inter2_kernel<<<B, 2 * d, shm, stream>>>(
                h1, adj, nodes, w2s, h2, N, d, K, B);
        }
    }

    // Final linear + log prior (wave per node)
    {
        const int wpb = 8;
        int grid = (B + wpb - 1) / wpb;
        final_linear_kernel<<<grid, 32 * wpb, 0, stream>>>(h2, lin_w, prior, out, B, d);
    }
}